// ManifoldRegularizer_79233556677071
// MI455X (gfx1250) — compile-verified
//
#include <hip/hip_runtime.h>
#include <hip/hip_bf16.h>
#include <math.h>

typedef __attribute__((ext_vector_type(2))) float v2f;
typedef __attribute__((ext_vector_type(8))) float v8f;

#define N 512
#define NSQ 8  // number of squarings

// C = (X^T X) * (1/t^2), X is [N x N] per matrix, one 64x64 C-tile per wave.
// For the first pass t_in == nullptr (scale 1); later passes X is symmetric so
// X^T X == X*X, which is exactly the squaring step.
__global__ __launch_bounds__(128)
void specnorm_xtx_kernel(const float* __restrict__ X, float* __restrict__ C,
                         const float* __restrict__ t_in) {
    const int b    = blockIdx.y;
    const int wave = threadIdx.x >> 5;            // 0..3
    const int lane = threadIdx.x & 31;
    const int tile = blockIdx.x * 4 + wave;       // 0..63 (8x8 tiles of 64x64)
    const int tm   = (tile >> 3) * 64;
    const int tn   = (tile & 7) * 64;

    const int half = lane >> 4;                   // K-half select (0: K=0,1  1: K=2,3)
    const int l16  = lane & 15;

    const float* __restrict__ Xb = X + (size_t)b * N * N;
    float* __restrict__ Cb       = C + (size_t)b * N * N;

    float inv_t2 = 1.0f;
    if (t_in) { float t = t_in[b]; inv_t2 = 1.0f / (t * t); }

    v8f acc[4][4];
#pragma unroll
    for (int mi = 0; mi < 4; ++mi)
#pragma unroll
        for (int ni = 0; ni < 4; ++ni)
            acc[mi][ni] = v8f{0.f,0.f,0.f,0.f,0.f,0.f,0.f,0.f};

    for (int k = 0; k < N; k += 4) {
        // A-frag: A_gemm[m][kk] = X[kk][m]; lane holds X[k+2*half + r][tm+mi*16+l16]
        // B-frag: B_gemm[kk][n] = X[kk][n]; lane holds X[k+2*half + r][tn+ni*16+l16]
        const float* __restrict__ rowp = Xb + (size_t)(k + 2 * half) * N;
        v2f af[4], bf[4];
#pragma unroll
        for (int mi = 0; mi < 4; ++mi) {
            const int col = tm + mi * 16 + l16;
            af[mi].x = rowp[col];
            af[mi].y = rowp[col + N];
        }
#pragma unroll
        for (int ni = 0; ni < 4; ++ni) {
            const int col = tn + ni * 16 + l16;
            bf[ni].x = rowp[col];
            bf[ni].y = rowp[col + N];
        }
#pragma unroll
        for (int mi = 0; mi < 4; ++mi)
#pragma unroll
            for (int ni = 0; ni < 4; ++ni)
                acc[mi][ni] = __builtin_amdgcn_wmma_f32_16x16x4_f32(
                    false, af[mi], false, bf[ni], (short)0, acc[mi][ni],
                    false, false);
    }

    // C/D layout: VGPR r -> row (tm + mi*16 + half*8 + r), col (tn + ni*16 + l16)
#pragma unroll
    for (int mi = 0; mi < 4; ++mi) {
#pragma unroll
        for (int ni = 0; ni < 4; ++ni) {
            const int col = tn + ni * 16 + l16;
#pragma unroll
            for (int r = 0; r < 8; ++r) {
                const int row = tm + mi * 16 + half * 8 + r;
                Cb[(size_t)row * N + col] = acc[mi][ni][r] * inv_t2;
            }
        }
    }
}

// t_out[b] = trace(C_b). One block per matrix; deterministic tree reduction.
__global__ __launch_bounds__(128)
void specnorm_trace_kernel(const float* __restrict__ C, float* __restrict__ t_out) {
    const int b   = blockIdx.x;
    const int tid = threadIdx.x;
    const float* __restrict__ Cb = C + (size_t)b * N * N;
    float s = 0.0f;
    for (int i = tid; i < N; i += 128) s += Cb[(size_t)i * N + i];
    __shared__ float red[128];
    red[tid] = s;
    __syncthreads();
    for (int off = 64; off > 0; off >>= 1) {
        if (tid < off) red[tid] += red[tid + off];
        __syncthreads();
    }
    if (tid == 0) t_out[b] = red[0];
}

// Unwind:  lambda1(M_i) = t_i * sqrt(lambda1(M_{i+1})),  lambda1(M_NSQ) ~= t_NSQ.
// sigma_b = sqrt(lambda1(M_0));   out = 0.01 * mean_b(sigma_b).
__global__ __launch_bounds__(128)
void specnorm_finalize_kernel(const float* __restrict__ tr, float* __restrict__ out,
                              int nb) {
    const int tid = threadIdx.x;
    float s = 0.0f;
    for (int b = tid; b < nb; b += 128) {
        float l = tr[(size_t)NSQ * nb + b];
        for (int i = NSQ - 1; i >= 0; --i)
            l = tr[(size_t)i * nb + b] * sqrtf(l);
        s += sqrtf(l);
    }
    __shared__ float red[128];
    red[tid] = s;
    __syncthreads();
    for (int off = 64; off > 0; off >>= 1) {
        if (tid < off) red[tid] += red[tid + off];
        __syncthreads();
    }
    if (tid == 0) out[0] = 0.01f * (red[0] / (float)nb);
}

extern "C" void kernel_launch(void* const* d_in, const int* in_sizes, int n_in,
                              void* d_out, int out_size, void* d_ws, size_t ws_size,
                              hipStream_t stream) {
    const float* A = (const float*)d_in[0];
    float* out     = (float*)d_out;
    const int nb   = in_sizes[0] / (N * N);   // 128

    const size_t MATBYTES = (size_t)nb * N * N * sizeof(float);  // 128 MB
    float* buf0 = (float*)d_ws;
    float* buf1 = (float*)((char*)d_ws + MATBYTES);
    float* tr   = (float*)((char*)d_ws + 2 * MATBYTES);          // (NSQ+1) * nb floats

    dim3 block(128);          // 4 waves, each one 64x64 tile
    dim3 grid(16, nb);        // 16 blocks * 4 waves = 64 tiles per matrix

    // M_0 = A^T A, t_0 = trace(M_0)
    specnorm_xtx_kernel<<<grid, block, 0, stream>>>(A, buf0, nullptr);
    specnorm_trace_kernel<<<nb, 128, 0, stream>>>(buf0, tr);

    // M_{i+1} = (M_i / t_i)^2, t_{i+1} = trace(M_{i+1})
    float* src = buf0;
    float* dst = buf1;
    for (int i = 0; i < NSQ; ++i) {
        specnorm_xtx_kernel<<<grid, block, 0, stream>>>(src, dst, tr + (size_t)i * nb);
        specnorm_trace_kernel<<<nb, 128, 0, stream>>>(dst, tr + (size_t)(i + 1) * nb);
        float* tmp = src; src = dst; dst = tmp;
    }

    specnorm_finalize_kernel<<<1, 128, 0, stream>>>(tr, out, nb);
}